// MessagePassing_9998683865750
// MI455X (gfx1250) — compile-verified
//
#include <hip/hip_runtime.h>

#define FEAT            128
#define EDGES_PER_WAVE  32
#define WAVES_PER_BLOCK 8
#define BLOCK_THREADS   (WAVES_PER_BLOCK * 32)
#define SCAN_THREADS    1024

typedef float v4f __attribute__((ext_vector_type(4)));

// ===========================================================================
// CSR "pull" path: build CSR by destination, then register-reduce per node.
// Roofline: x (25.6MB) and the CSR (2.6MB) live in the 192MB L2; the gather
// reads 328MB from L2, writes 25.6MB NT to HBM. No fp atomics anywhere.
// ===========================================================================

// --- K1: zero the per-node counters --------------------------------------
__global__ void zero_int_kernel(int* __restrict__ p, int n) {
    int i = blockIdx.x * blockDim.x + threadIdx.x;
    if (i < n) p[i] = 0;
}

// --- K2: histogram of destination indices (no-return u32 atomics) --------
__global__ __launch_bounds__(BLOCK_THREADS)
void count_kernel(const long long* __restrict__ ei, int* __restrict__ cnt, int E) {
    int e = blockIdx.x * blockDim.x + threadIdx.x;
    if (e < E) {
        int d = (int)ei[(long long)E + e];           // dst = edge_index[1][e]
        (void)__hip_atomic_fetch_add(cnt + d, 1, __ATOMIC_RELAXED,
                                     __HIP_MEMORY_SCOPE_AGENT);
    }
}

// --- K3: single-block exclusive scan (chunk-serial + one LDS block scan) --
__global__ __launch_bounds__(SCAN_THREADS)
void scan_kernel(int* __restrict__ cnt, int* __restrict__ off, int N) {
    __shared__ int sm[SCAN_THREADS];
    const int tid = threadIdx.x;
    const int C   = (N + SCAN_THREADS - 1) / SCAN_THREADS;
    const int b   = tid * C;
    const int e   = min(N, b + C);

    int sum = 0;
    for (int i = b; i < e; ++i) sum += cnt[i];

    sm[tid] = sum;
    __syncthreads();
    for (int ofs = 1; ofs < SCAN_THREADS; ofs <<= 1) {
        int t = (tid >= ofs) ? sm[tid - ofs] : 0;
        __syncthreads();
        sm[tid] += t;
        __syncthreads();
    }
    int run = sm[tid] - sum;                          // exclusive base
    for (int i = b; i < e; ++i) {
        int v = cnt[i];
        off[i] = run;
        cnt[i] = run;                                 // cursor for fill pass
        run += v;
    }
    if (tid == SCAN_THREADS - 1) off[N] = run;        // == E
}

// --- K4: scatter edges into CSR slots (returning u32 atomics) ------------
__global__ __launch_bounds__(BLOCK_THREADS)
void fill_kernel(const long long* __restrict__ ei, int* __restrict__ cur,
                 int* __restrict__ csr, int E) {
    int e = blockIdx.x * blockDim.x + threadIdx.x;
    if (e < E) {
        int s = (int)ei[e];
        int d = (int)ei[(long long)E + e];
        int p = __hip_atomic_fetch_add(cur + d, 1, __ATOMIC_RELAXED,
                                       __HIP_MEMORY_SCOPE_AGENT);
        csr[p] = s;
    }
}

// --- K5: per-node register reduction -------------------------------------
// Wave per node; lane l owns features [4l..4l+3].
//  * The node's CSR index list is staged into a private 128B LDS slot with
//    per-lane global_load_async_to_lds_b32 (ASYNCcnt path) -> keeps the
//    LOADcnt queue free for the latency-critical row gathers.
//  * Indices re-read as uniform ds_load broadcasts + readfirstlane -> SGPR,
//    so the b128 row gathers use saddr form.
//  * 4 independent global_load_b128 in flight per iteration; accumulate in
//    VGPRs; one non-temporal b128 store per lane (out never re-read).
__global__ __launch_bounds__(BLOCK_THREADS)
void gather_kernel(const float* __restrict__ x, const int* __restrict__ off,
                   const int* __restrict__ csr, float* __restrict__ out,
                   int N, int E) {
    __shared__ int sidx[WAVES_PER_BLOCK][32];

    const int lane = threadIdx.x & 31;
    const int wv   = threadIdx.x >> 5;
    const int node = blockIdx.x * WAVES_PER_BLOCK + wv;
    if (node >= N) return;                            // uniform per wave

    const int beg = off[node];
    const int end = off[node + 1];

    const v4f* __restrict__ x4 = (const v4f*)x;       // 32 float4 per row
    v4f acc = (v4f)0.0f;

    // LDS byte offset of this lane's staging slot (low 32 bits of the
    // generic pointer to LDS == wave-relative LDS byte address).
    const unsigned ldsAddr = (unsigned)(size_t)(void*)&sidx[wv][lane];

    for (int cbase = beg; cbase < end; cbase += 32) {
        // ---- async stage up to 32 indices into LDS (clamped, in-bounds) --
        int j = cbase + lane;
        if (j > E - 1) j = E - 1;                     // safe: extra ignored
        const unsigned long long ga = (unsigned long long)(size_t)(csr + j);
        asm volatile("global_load_async_to_lds_b32 %0, %1, off"
                     :: "v"(ldsAddr), "v"(ga) : "memory");
        asm volatile("s_wait_asynccnt 0x0" ::: "memory");

        const int m = min(32, end - cbase);
        int k = 0;
        for (; k + 4 <= m; k += 4) {
            const int s0 = __builtin_amdgcn_readfirstlane(sidx[wv][k + 0]);
            const int s1 = __builtin_amdgcn_readfirstlane(sidx[wv][k + 1]);
            const int s2 = __builtin_amdgcn_readfirstlane(sidx[wv][k + 2]);
            const int s3 = __builtin_amdgcn_readfirstlane(sidx[wv][k + 3]);
            const v4f a = x4[(size_t)s0 * 32 + lane]; // 4 independent
            const v4f b = x4[(size_t)s1 * 32 + lane]; // saddr-form
            const v4f c = x4[(size_t)s2 * 32 + lane]; // global_load_b128
            const v4f d = x4[(size_t)s3 * 32 + lane]; // in flight
            acc += (a + b) + (c + d);
        }
        for (; k < m; ++k) {
            const int s = __builtin_amdgcn_readfirstlane(sidx[wv][k]);
            acc += x4[(size_t)s * 32 + lane];
        }
    }

    // Streamed once, never re-read: NT store keeps L2 dedicated to x.
    __builtin_nontemporal_store(acc, (v4f*)out + (size_t)node * 32 + lane);
}

// ===========================================================================
// Fallback "push" path (atomic scatter-add) if the workspace is too small.
// ===========================================================================
__global__ void zero_out_kernel(v4f* __restrict__ out, int n4) {
    int i = blockIdx.x * blockDim.x + threadIdx.x;
    if (i < n4) out[i] = (v4f)0.0f;
}

__global__ __launch_bounds__(BLOCK_THREADS)
void scatter_add_kernel(const float* __restrict__ x,
                        const long long* __restrict__ ei,
                        float* __restrict__ out, int E) {
    const int lane = threadIdx.x & 31;
    const int wave = blockIdx.x * WAVES_PER_BLOCK + (threadIdx.x >> 5);
    const long long base = (long long)wave * EDGES_PER_WAVE;
    if (base >= (long long)E) return;
    const int cnt = (int)min((long long)EDGES_PER_WAVE, (long long)E - base);

    const long long eIdx = base + lane;
    int s_l = 0, d_l = 0;
    if (eIdx < (long long)E) {
        s_l = (int)ei[eIdx];
        d_l = (int)ei[(long long)E + eIdx];
    }
    const v4f* __restrict__ x4 = (const v4f*)x;
    for (int e = 0; e < cnt; ++e) {
        const int s = __builtin_amdgcn_readlane(s_l, e);
        const int d = __builtin_amdgcn_readlane(d_l, e);
        if (e + 1 < cnt) {
            const int sn = __builtin_amdgcn_readlane(s_l, e + 1);
            __builtin_prefetch((const void*)(x4 + (size_t)sn * 32 + lane), 0, 1);
        }
        const v4f v = x4[(size_t)s * 32 + lane];
        float* o = out + (size_t)d * FEAT + (lane << 2);
        (void)__hip_atomic_fetch_add(o + 0, v.x, __ATOMIC_RELAXED, __HIP_MEMORY_SCOPE_AGENT);
        (void)__hip_atomic_fetch_add(o + 1, v.y, __ATOMIC_RELAXED, __HIP_MEMORY_SCOPE_AGENT);
        (void)__hip_atomic_fetch_add(o + 2, v.z, __ATOMIC_RELAXED, __HIP_MEMORY_SCOPE_AGENT);
        (void)__hip_atomic_fetch_add(o + 3, v.w, __ATOMIC_RELAXED, __HIP_MEMORY_SCOPE_AGENT);
    }
}

// ===========================================================================
// Launch
// ===========================================================================
extern "C" void kernel_launch(void* const* d_in, const int* in_sizes, int n_in,
                              void* d_out, int out_size, void* d_ws, size_t ws_size,
                              hipStream_t stream) {
    const float*     x   = (const float*)d_in[0];
    const long long* ei  = (const long long*)d_in[1];
    float*           out = (float*)d_out;

    const int E = in_sizes[1] / 2;                    // 640000
    const int N = out_size / FEAT;                    // 50000

    const size_t need = (size_t)(N + 1 + N + E) * sizeof(int);
    if (ws_size >= need) {
        int* off = (int*)d_ws;                        // N+1
        int* cur = off + (N + 1);                     // N
        int* csr = cur + N;                           // E

        zero_int_kernel<<<(N + BLOCK_THREADS - 1) / BLOCK_THREADS,
                          BLOCK_THREADS, 0, stream>>>(cur, N);
        count_kernel<<<(E + BLOCK_THREADS - 1) / BLOCK_THREADS,
                       BLOCK_THREADS, 0, stream>>>(ei, cur, E);
        scan_kernel<<<1, SCAN_THREADS, 0, stream>>>(cur, off, N);
        fill_kernel<<<(E + BLOCK_THREADS - 1) / BLOCK_THREADS,
                      BLOCK_THREADS, 0, stream>>>(ei, cur, csr, E);
        const int gblocks = (N + WAVES_PER_BLOCK - 1) / WAVES_PER_BLOCK;
        gather_kernel<<<gblocks, BLOCK_THREADS, 0, stream>>>(x, off, csr, out, N, E);
    } else {
        const int n4 = out_size / 4;
        zero_out_kernel<<<(n4 + BLOCK_THREADS - 1) / BLOCK_THREADS,
                          BLOCK_THREADS, 0, stream>>>((v4f*)out, n4);
        const int waves  = (E + EDGES_PER_WAVE - 1) / EDGES_PER_WAVE;
        const int blocks = (waves + WAVES_PER_BLOCK - 1) / WAVES_PER_BLOCK;
        scatter_add_kernel<<<blocks, BLOCK_THREADS, 0, stream>>>(x, ei, out, E);
    }
}